// EarlyRetiermentBlock_24567212933436
// MI455X (gfx1250) — compile-verified
//
#include <hip/hip_runtime.h>
#include <math.h>

// ---------------- constants ----------------
#define YEAR_COLS 37
#define NINW 40          // work-block input width
#define NCOLS 42         // 40 work cols + b_t + b_bar
#define HW 10
#define HR 5
#define NCHUNK 10        // 40 / 4 K-chunks for f32 WMMA

// ws float layout: [0..41] sums, [42..83] sumsq, [84..123] scale, [124..163] shift,
// [164..169] retire (aS,aT,btS,btT,bbS,bbT)
#define WS_SUM 0
#define WS_SQ 42
#define WS_SC 84
#define WS_SH 124
#define WS_RET 164

typedef float v2f __attribute__((ext_vector_type(2)));
typedef float v8f __attribute__((ext_vector_type(8)));

__device__ __forceinline__ float gelu_f(float x) {
  // exact GELU: x * 0.5 * (1 + erf(x / sqrt(2)))
  return 0.5f * x * (1.0f + erff(x * 0.70710678118654752440f));
}
__device__ __forceinline__ float sigmoid_f(float x) {
  return 1.0f / (1.0f + __expf(-x));
}

__device__ __forceinline__ float load_col(int c, int row,
    const float* __restrict__ theta, const float* __restrict__ edu,
    const float* __restrict__ a_t, const float* __restrict__ all_y) {
  if (c == 0) return theta[row];
  if (c == 1) return edu[row];
  if (c == 2) return a_t[row];
  return all_y[(size_t)row * YEAR_COLS + (c - 3)];
}

// ---------------- kernel 0: zero the accumulator region ----------------
__global__ void zero_ws_kernel(float* __restrict__ ws) {
  int t = threadIdx.x;
  if (t < 2 * NCOLS) ws[t] = 0.0f;
}

// ---------------- kernel 1: column sums / sums-of-squares ----------------
__global__ __launch_bounds__(256)
void stats_kernel(const float* __restrict__ theta, const float* __restrict__ edu,
                  const float* __restrict__ a_t, const float* __restrict__ all_y,
                  const float* __restrict__ b_t, const float* __restrict__ b_bar,
                  float* __restrict__ ws, int B) {
  float s[NCOLS], q[NCOLS];
#pragma unroll
  for (int c = 0; c < NCOLS; ++c) { s[c] = 0.0f; q[c] = 0.0f; }

  int stride = gridDim.x * blockDim.x;
  for (int i = blockIdx.x * blockDim.x + threadIdx.x; i < B; i += stride) {
    float v;
    v = theta[i]; s[0] += v; q[0] += v * v;
    v = edu[i];   s[1] += v; q[1] += v * v;
    v = a_t[i];   s[2] += v; q[2] += v * v;
    const float* __restrict__ yr = all_y + (size_t)i * YEAR_COLS;
#pragma unroll
    for (int j = 0; j < YEAR_COLS; ++j) { v = yr[j]; s[3 + j] += v; q[3 + j] += v * v; }
    v = b_t[i];   s[40] += v; q[40] += v * v;
    v = b_bar[i]; s[41] += v; q[41] += v * v;
  }

  // intra-wave reduction (wave32)
#pragma unroll
  for (int c = 0; c < NCOLS; ++c) {
#pragma unroll
    for (int off = 16; off > 0; off >>= 1) {
      s[c] += __shfl_down(s[c], off, 32);
      q[c] += __shfl_down(q[c], off, 32);
    }
  }

  __shared__ float red[2 * NCOLS];
  for (int t = threadIdx.x; t < 2 * NCOLS; t += blockDim.x) red[t] = 0.0f;
  __syncthreads();
  if ((threadIdx.x & 31) == 0) {
#pragma unroll
    for (int c = 0; c < NCOLS; ++c) {
      atomicAdd(&red[c], s[c]);
      atomicAdd(&red[NCOLS + c], q[c]);
    }
  }
  __syncthreads();
  for (int t = threadIdx.x; t < 2 * NCOLS; t += blockDim.x) atomicAdd(&ws[t], red[t]);
}

// ---------------- kernel 2: fold mean/var/gamma/beta -> scale/shift ----------------
__global__ void finalize_kernel(const float* __restrict__ w_bn_g, const float* __restrict__ w_bn_b,
                                const float* __restrict__ r_bn_g, const float* __restrict__ r_bn_b,
                                float* __restrict__ ws, int B) {
  int c = threadIdx.x;
  if (c >= NCOLS) return;
  float invB = 1.0f / (float)B;
  float mean = ws[WS_SUM + c] * invB;
  float var = ws[WS_SQ + c] * invB - mean * mean;
  float inv = rsqrtf(var + 1e-5f);
  if (c < NINW) {
    float sc = inv * w_bn_g[c];
    ws[WS_SC + c] = sc;
    ws[WS_SH + c] = w_bn_b[c] - mean * sc;
  }
  if (c == 2)  { float sc = inv * r_bn_g[0]; ws[WS_RET + 0] = sc; ws[WS_RET + 1] = r_bn_b[0] - mean * sc; }
  if (c == 40) { float sc = inv * r_bn_g[1]; ws[WS_RET + 2] = sc; ws[WS_RET + 3] = r_bn_b[1] - mean * sc; }
  if (c == 41) { float sc = inv * r_bn_g[1]; ws[WS_RET + 4] = sc; ws[WS_RET + 5] = r_bn_b[1] - mean * sc; }
}

// ---------------- kernel 3: fused forward ----------------
// Phase A: per-wave 16-row tiles, x[16,40] @ w_g1[40,10] via V_WMMA_F32_16X16X4_F32.
//   B-fragments (w_g1) and BN scale/shift are hoisted into registers once per wave;
//   N>=10 columns are left as don't-care (never read), so no padding branches.
// Phase B: one element per lane, remaining tiny matmuls + heads + policy math.
__global__ __launch_bounds__(256)
void fused_kernel(const float* __restrict__ theta, const float* __restrict__ edu,
                  const float* __restrict__ a_t, const float* __restrict__ all_y,
                  const float* __restrict__ w_t, const float* __restrict__ b_t,
                  const float* __restrict__ pr_bar, const float* __restrict__ b_bar,
                  const float* __restrict__ w_g1, const float* __restrict__ w_g1b,
                  const float* __restrict__ w_g2, const float* __restrict__ w_g2b,
                  const float* __restrict__ w_th, const float* __restrict__ w_thb,
                  const float* __restrict__ w_ta, const float* __restrict__ w_tab,
                  const float* __restrict__ w_yh, const float* __restrict__ w_yhb,
                  const float* __restrict__ w_ya, const float* __restrict__ w_yab,
                  const float* __restrict__ r_l1, const float* __restrict__ r_l1b,
                  const float* __restrict__ r_l2, const float* __restrict__ r_l2b,
                  const float* __restrict__ p_w, const float* __restrict__ p_b,
                  const float* __restrict__ ws, float* __restrict__ out, int B) {
  __shared__ float s_scale[NINW];
  __shared__ float s_shift[NINW];
  __shared__ float s_xx[256 * 13];   // stride 13: conflict-free row reads in phase B

  if (threadIdx.x < NINW) {
    s_scale[threadIdx.x] = ws[WS_SC + threadIdx.x];
    s_shift[threadIdx.x] = ws[WS_SH + threadIdx.x];
  }
  __syncthreads();

  const int wave = threadIdx.x >> 5;
  const int lane = threadIdx.x & 31;
  const int half = lane >> 4;   // K-pair selector for A-frag; M-half for C-frag
  const int l16  = lane & 15;   // M (rows of A) and N (cols of B/C)
  const int base = blockIdx.x * 256;

  // ---- hoisted per-lane invariants: B-fragments of w_g1 and BN scale/shift pairs ----
  const int ncl = (l16 < HW) ? l16 : (HW - 1);   // clamp: N>=10 outputs are never read
  v2f Bfrag[NCHUNK];
  v2f scp[NCHUNK], shp[NCHUNK];
#pragma unroll
  for (int c = 0; c < NCHUNK; ++c) {
    const int ka = c * 4 + half * 2;
    Bfrag[c].x = w_g1[ka * HW + ncl];
    Bfrag[c].y = w_g1[(ka + 1) * HW + ncl];
    scp[c].x = s_scale[ka];
    scp[c].y = s_scale[ka + 1];
    shp[c].x = s_shift[ka];
    shp[c].y = s_shift[ka + 1];
  }
  const float bias = (l16 < HW) ? w_g1b[ncl] : 0.0f;

#pragma unroll
  for (int s = 0; s < 2; ++s) {
    const int rowbase = base + wave * 32 + s * 16;
    int row = rowbase + l16;
    if (row >= B) row = B - 1;           // clamp; garbage rows masked at store time

    v8f acc = {0.f, 0.f, 0.f, 0.f, 0.f, 0.f, 0.f, 0.f};
#pragma unroll
    for (int c = 0; c < NCHUNK; ++c) {
      const int ka = c * 4 + half * 2;
      // A fragment (16x4 f32): lane l16 = row M; VGPR0/1 = K pair (ka, ka+1)
      v2f xv;
      xv.x = load_col(ka, row, theta, edu, a_t, all_y);
      xv.y = load_col(ka + 1, row, theta, edu, a_t, all_y);
      v2f A = xv * scp[c] + shp[c];   // packed FMA normalize
      acc = __builtin_amdgcn_wmma_f32_16x16x4_f32(false, A, false, Bfrag[c],
                                                  (short)0, acc, false, false);
    }
    // C layout: lane l16 = N, VGPR j = M (j or j+8 by lane half). Bias + GELU + LDS.
    if (l16 < HW) {
#pragma unroll
      for (int j = 0; j < 8; ++j) {
        const int m = j + half * 8;
        s_xx[(wave * 32 + s * 16 + m) * 13 + l16] = gelu_f(acc[j] + bias);
      }
    }
  }
  __syncthreads();

  // ---- Phase B: one element per thread ----
  const int i = base + threadIdx.x;
  if (i >= B) return;

  float xxv[HW];
#pragma unroll
  for (int k = 0; k < HW; ++k) xxv[k] = s_xx[threadIdx.x * 13 + k];

  // xx_w = gelu(xx @ w_g2 + b)
  float xx[HW];
#pragma unroll
  for (int n = 0; n < HW; ++n) {
    float t = w_g2b[n];
#pragma unroll
    for (int k = 0; k < HW; ++k) t += xxv[k] * w_g2[k * HW + n];
    xx[n] = gelu_f(t);
  }

  // x_h / x_a heads
  float xh[HW], xa[HW];
#pragma unroll
  for (int n = 0; n < HW; ++n) {
    float th = w_thb[n], ta = w_tab[n];
#pragma unroll
    for (int k = 0; k < HW; ++k) {
      th += xx[k] * w_th[k * HW + n];
      ta += xx[k] * w_ta[k * HW + n];
    }
    xh[n] = gelu_f(th);
    xa[n] = gelu_f(ta);
  }

  // softmax head -> h_t
  float z[4];
#pragma unroll
  for (int j = 0; j < 4; ++j) {
    float t = w_yhb[j];
#pragma unroll
    for (int k = 0; k < HW; ++k) t += xh[k] * w_yh[k * 4 + j];
    z[j] = t;
  }
  float mz = fmaxf(fmaxf(z[0], z[1]), fmaxf(z[2], z[3]));
  float e0 = __expf(z[0] - mz), e1 = __expf(z[1] - mz);
  float e2 = __expf(z[2] - mz), e3 = __expf(z[3] - mz);
  float h_t = (e1 * 1300.0f + e2 * 2080.0f + e3 * 2860.0f) / (e0 + e1 + e2 + e3);

  // sigmoid head -> x_w
  float sw = w_yab[0];
#pragma unroll
  for (int k = 0; k < HW; ++k) sw += xa[k] * w_ya[k];
  float x_w = sigmoid_f(sw);

  // retirement block (shared weights, two BN'd b inputs)
  const float a  = a_t[i];
  const float bt = b_t[i];
  const float bb = b_bar[i];
  const float an  = a  * ws[WS_RET + 0] + ws[WS_RET + 1];
  const float bn1 = bt * ws[WS_RET + 2] + ws[WS_RET + 3];
  const float bn2 = bb * ws[WS_RET + 4] + ws[WS_RET + 5];
  float xr_pre[HR], xrr_pre[HR];
#pragma unroll
  for (int j = 0; j < HR; ++j) {
    const float w0 = r_l1[j], w1 = r_l1[HR + j], bj = r_l1b[j];
    xr_pre[j]  = an * w0 + bn1 * w1 + bj;
    xrr_pre[j] = an * w0 + bn2 * w1 + bj;
  }
  float sr = r_l2b[0], srr = r_l2b[0];
#pragma unroll
  for (int j = 0; j < HR; ++j) {
    sr  += gelu_f(xr_pre[j])  * r_l2[j];
    srr += gelu_f(xrr_pre[j]) * r_l2[j];
  }
  const float x_r  = sigmoid_f(sr);
  const float x_rr = sigmoid_f(srr);

  // pr head: sigmoid([xx_w, xx_r_pre] @ p_w + p_b)
  float pl = p_b[0];
#pragma unroll
  for (int k = 0; k < HW; ++k) pl += xx[k] * p_w[k];
#pragma unroll
  for (int j = 0; j < HR; ++j) pl += xr_pre[j] * p_w[HW + j];
  const float pr = sigmoid_f(pl);

  // economic policy math
  const float wt  = w_t[i];
  const float prb = pr_bar[i];
  const float y    = wt * h_t;
  const float netw = y - 0.062f * fminf(y, 147000.0f) + a;
  const float c_w  = x_w * netw + 1e-8f;
  const float a_w  = (1.0f - x_w) * netw * 1.02f;
  const float ab   = bt + a;
  const float a_r  = (1.0f - x_r) * ab * 1.02f;
  const float c_r  = x_r * ab + 1e-8f;
  const float abb  = bb + a;
  const float a_rr = (1.0f - x_rr) * abb * 1.02f;
  const float c_rr = x_rr * abb + 1e-8f;
  const float om   = 1.0f - prb;
  const float c_t   = om * (pr * c_r + (1.0f - pr) * c_w) + prb * c_rr;
  const float a_tp  = om * (pr * a_r + (1.0f - pr) * a_w) + prb * a_rr;
  const float prbtp = prb + om * pr;
  const float bbtp  = prb * bb + om * bt;

  const size_t Bs = (size_t)B;
  out[ 0 * Bs + i] = a_tp;
  out[ 1 * Bs + i] = c_t;
  out[ 2 * Bs + i] = a_w;
  out[ 3 * Bs + i] = c_w;
  out[ 4 * Bs + i] = a_r;
  out[ 5 * Bs + i] = c_r;
  out[ 6 * Bs + i] = a_rr;
  out[ 7 * Bs + i] = c_rr;
  out[ 8 * Bs + i] = y;
  out[ 9 * Bs + i] = h_t;
  out[10 * Bs + i] = pr;
  out[11 * Bs + i] = prbtp;
  out[12 * Bs + i] = bbtp;
}

// ---------------- host launcher ----------------
extern "C" void kernel_launch(void* const* d_in, const int* in_sizes, int n_in,
                              void* d_out, int out_size, void* d_ws, size_t ws_size,
                              hipStream_t stream) {
  // inputs: theta, edu, a_t, all_y, w_t, b_t, pr_bar_t, b_bar_t, then 22 param leaves
  const float* theta  = (const float*)d_in[0];
  const float* edu    = (const float*)d_in[1];
  const float* a_t    = (const float*)d_in[2];
  const float* all_y  = (const float*)d_in[3];
  const float* w_t    = (const float*)d_in[4];
  const float* b_t    = (const float*)d_in[5];
  const float* pr_bar = (const float*)d_in[6];
  const float* b_bar  = (const float*)d_in[7];
  const float* w_bn_g = (const float*)d_in[8];
  const float* w_bn_b = (const float*)d_in[9];
  const float* w_g1   = (const float*)d_in[10];
  const float* w_g1b  = (const float*)d_in[11];
  const float* w_g2   = (const float*)d_in[12];
  const float* w_g2b  = (const float*)d_in[13];
  const float* w_th   = (const float*)d_in[14];
  const float* w_thb  = (const float*)d_in[15];
  const float* w_ta   = (const float*)d_in[16];
  const float* w_tab  = (const float*)d_in[17];
  const float* w_yh   = (const float*)d_in[18];
  const float* w_yhb  = (const float*)d_in[19];
  const float* w_ya   = (const float*)d_in[20];
  const float* w_yab  = (const float*)d_in[21];
  const float* r_bn_g = (const float*)d_in[22];
  const float* r_bn_b = (const float*)d_in[23];
  const float* r_l1   = (const float*)d_in[24];
  const float* r_l1b  = (const float*)d_in[25];
  const float* r_l2   = (const float*)d_in[26];
  const float* r_l2b  = (const float*)d_in[27];
  const float* p_w    = (const float*)d_in[28];
  const float* p_b    = (const float*)d_in[29];

  const int B = in_sizes[0];
  float* ws = (float*)d_ws;
  float* outp = (float*)d_out;

  zero_ws_kernel<<<1, 256, 0, stream>>>(ws);
  stats_kernel<<<512, 256, 0, stream>>>(theta, edu, a_t, all_y, b_t, b_bar, ws, B);
  finalize_kernel<<<1, 64, 0, stream>>>(w_bn_g, w_bn_b, r_bn_g, r_bn_b, ws, B);

  const int nb = (B + 255) / 256;
  fused_kernel<<<nb, 256, 0, stream>>>(theta, edu, a_t, all_y, w_t, b_t, pr_bar, b_bar,
                                       w_g1, w_g1b, w_g2, w_g2b, w_th, w_thb, w_ta, w_tab,
                                       w_yh, w_yhb, w_ya, w_yab, r_l1, r_l1b, r_l2, r_l2b,
                                       p_w, p_b, ws, outp, B);
}